// InvariantGaussianAttention_2602750182102
// MI455X (gfx1250) — compile-verified
//
#include <hip/hip_runtime.h>
#include <math.h>

// ---------------- problem constants ----------------
#define B_   2
#define N_   384
#define CS_  384
#define CZ_  128
#define H_   8
#define CH_  16
#define G_   2
#define P_   8
#define CZ4_ 32
#define NT16 (N_ / 16)          // 24 j/i tiles
#define HC_  (H_ * CH_)         // 128
#define CATD 640                // H*(CH + P*4 + CZ4)

typedef __attribute__((ext_vector_type(2))) float v2f;
typedef __attribute__((ext_vector_type(8))) float v8f;
typedef __attribute__((ext_vector_type(4))) int   v4i;

// D = A(16x4,f32) * B(4x16,f32) + C(16x16,f32)
__device__ __forceinline__ v8f wmma4(v2f a, v2f b, v8f c) {
  return __builtin_amdgcn_wmma_f32_16x16x4_f32(false, a, false, b, (short)0, c,
                                               false, false);
}

// ---- CDNA5 async global->LDS staging (guarded; falls back to b128+ds) ----
#if defined(__AMDGCN__) && __has_builtin(__builtin_amdgcn_global_load_async_to_lds_b128)
#define IGA_ASYNC_LDS 1
#else
#define IGA_ASYNC_LDS 0
#endif

#if IGA_ASYNC_LDS
__device__ __forceinline__ void iga_async_b128(const float* g, float* l) {
  // builtin signature (probe-confirmed): (int4 addrspace(1)*, int4 addrspace(3)*, imm, imm)
  __builtin_amdgcn_global_load_async_to_lds_b128(
      (__attribute__((address_space(1))) v4i*)g,
      (__attribute__((address_space(3))) v4i*)l, 0, 0);
}
__device__ __forceinline__ void iga_wait_async0() {
#if __has_builtin(__builtin_amdgcn_s_wait_asynccnt)
  __builtin_amdgcn_s_wait_asynccnt(0);
#else
  asm volatile("s_wait_asynccnt 0x0" ::: "memory");
#endif
}
#endif

// ---------------------------------------------------------------------------
// Generic f32 WMMA GEMM:  C[M x Nc] = A[M x K] @ W[K x Nc] (+ bias[Nc])
// One wave per 16x16 output tile. A,W row-major f32.
// A frag (16x4): lane L -> row M=L&15, K = kb + 2*(L>>4) + {0,1}
// B frag (4x16): lane L -> col N=L&15, K = kb + 2*(L>>4) + {0,1}
// C frag: VGPR r, lane L -> (M = r + 8*(L>>4), N = L&15)
// ---------------------------------------------------------------------------
__global__ __launch_bounds__(32) void gemm_f32_wmma(
    const float* __restrict__ A, const float* __restrict__ W,
    const float* __restrict__ bias, float* __restrict__ C, int M, int K,
    int Nc) {
  const int L = threadIdx.x, lo = L & 15, hi = L >> 4;
  const int mb = blockIdx.x * 16;
  const int nb = blockIdx.y * 16;
  const float* Arow = A + (size_t)(mb + lo) * K;
  const int col = nb + lo;
  v8f acc = {};
  for (int k = 0; k < K; k += 4) {
    const int ka = k + 2 * hi;
    v2f a; a.x = Arow[ka]; a.y = Arow[ka + 1];
    v2f b; b.x = W[(size_t)ka * Nc + col]; b.y = W[(size_t)(ka + 1) * Nc + col];
    acc = wmma4(a, b, acc);
  }
  const float bv = bias ? bias[col] : 0.0f;
#pragma unroll
  for (int r = 0; r < 8; ++r)
    C[(size_t)(mb + r + 8 * hi) * Nc + col] = acc[r] + bv;
}

// ---------------------------------------------------------------------------
// Gaussian prep: per (b,n,h,g) build global mu(3) and symmetric Sig(6) for
// both q and k sides.  Sig = R diag(var) R^T ; mu = R pos + t.
// ---------------------------------------------------------------------------
__global__ void iga_geom_prep(const float* __restrict__ qg,
                              const float* __restrict__ kg,
                              const float* __restrict__ lmean,
                              const float* __restrict__ slog,
                              const float* __restrict__ rot,
                              const float* __restrict__ trans,
                              float* __restrict__ qmu, float* __restrict__ qsig,
                              float* __restrict__ kmu,
                              float* __restrict__ ksig) {
  const int idx = blockIdx.x * blockDim.x + threadIdx.x;
  if (idx >= B_ * N_ * H_ * G_) return;
  const int g = idx & (G_ - 1);
  const int h = (idx / G_) & (H_ - 1);
  const int n = (idx / (G_ * H_)) % N_;
  const int b = idx / (G_ * H_ * N_);
  const int bn = b * N_ + n;
  const float* R = rot + (size_t)bn * 9;
  const float* t = trans + (size_t)bn * 3;
  const float* am = lmean + (size_t)bn * 3;
  const float* bs = slog + (size_t)bn * 3;
#pragma unroll
  for (int side = 0; side < 2; ++side) {
    const float* feat =
        (side ? kg : qg) + (size_t)bn * (H_ * G_ * 6) + (h * G_ + g) * 6;
    float pos[3], var[3];
#pragma unroll
    for (int c = 0; c < 3; ++c) {
      const float u = tanhf(feat[c]) * 3.0f;        // UMAX
      const float sld = tanhf(feat[3 + c]) * 2.0f;  // LMAX
      const float e = bs[c] + sld;
      const float sig = fmaxf(__expf(e), 1e-6f);
      pos[c] = am[c] + u * sig;
      var[c] = __expf(2.0f * e);
    }
    float* MU = (side ? kmu : qmu) + (size_t)idx * 3;
    float* SG = (side ? ksig : qsig) + (size_t)idx * 6;
#pragma unroll
    for (int i = 0; i < 3; ++i)
      MU[i] = R[i * 3] * pos[0] + R[i * 3 + 1] * pos[1] + R[i * 3 + 2] * pos[2] +
              t[i];
    int c6 = 0;
#pragma unroll
    for (int i = 0; i < 3; ++i)
#pragma unroll
      for (int k2 = 0; k2 < 3; ++k2) {
        if (k2 < i) continue;
        SG[c6++] = R[i * 3] * var[0] * R[k2 * 3] +
                   R[i * 3 + 1] * var[1] * R[k2 * 3 + 1] +
                   R[i * 3 + 2] * var[2] * R[k2 * 3 + 2];
      }
  }
}

// value points -> global frame: vpg[(bn*H+h)*24 + p*3 + c]
__global__ void iga_vpg_prep(const float* __restrict__ vpts,
                             const float* __restrict__ rot,
                             const float* __restrict__ trans,
                             float* __restrict__ vpg) {
  const int idx = blockIdx.x * blockDim.x + threadIdx.x;
  if (idx >= B_ * N_ * H_ * P_) return;
  const int p = idx & (P_ - 1);
  const int h = (idx / P_) & (H_ - 1);
  const int n = (idx / (P_ * H_)) % N_;
  const int b = idx / (P_ * H_ * N_);
  const int bn = b * N_ + n;
  const float* R = rot + (size_t)bn * 9;
  const float* t = trans + (size_t)bn * 3;
  const float* v = vpts + (size_t)bn * (H_ * P_ * 3) + (h * P_ + p) * 3;
  float* o = vpg + ((size_t)bn * H_ + h) * (P_ * 3) + p * 3;
#pragma unroll
  for (int i = 0; i < 3; ++i)
    o[i] = R[i * 3] * v[0] + R[i * 3 + 1] * v[1] + R[i * 3 + 2] * v[2] + t[i];
}

// ---------------------------------------------------------------------------
// Fused z projections: single streaming pass over z (the 150MB tensor — the
// roofline-dominant stream).  Each wave stages a 16x128 f32 tile of z into
// LDS with GLOBAL_LOAD_ASYNC_TO_LDS_B128 (b128/lane, ASYNCcnt tracked), then
// runs 3 WMMA tiles per k-step: pz cols 0-15, pz cols 16-31, Wb (8 cols,
// zero-padded to 16).  LDS row pad of 4 dwords makes the A-frag ds reads
// bank-conflict-free (dword bank = 4*lo + 2*hi + {0,1}).
// ---------------------------------------------------------------------------
__global__ __launch_bounds__(32) void iga_zproj(
    const float* __restrict__ z, const float* __restrict__ Wdz,
    const float* __restrict__ bdz, const float* __restrict__ Wb,
    const float* __restrict__ bb, float* __restrict__ pz,
    float* __restrict__ bias2d) {
  const int L = threadIdx.x, lo = L & 15, hi = L >> 4;
  const size_t rowb = (size_t)blockIdx.x * 16;
  __shared__ float sZ[16][CZ_ + 4];

  // stage: issue t covers row t (128 floats = 32 float4 = one per lane)
#if IGA_ASYNC_LDS
#pragma unroll
  for (int t = 0; t < 16; ++t)
    iga_async_b128(z + (rowb + t) * CZ_ + L * 4, &sZ[t][L * 4]);
  iga_wait_async0();
#else
#pragma unroll
  for (int t = 0; t < 16; ++t)
    *(float4*)&sZ[t][L * 4] = *(const float4*)(z + (rowb + t) * CZ_ + L * 4);
  __syncthreads();
#endif

  v8f c0 = {}, c1 = {}, cb = {};
  for (int k = 0; k < CZ_; k += 4) {
    const int ka = k + 2 * hi;
    v2f a; a.x = sZ[lo][ka]; a.y = sZ[lo][ka + 1];
    v2f b0; b0.x = Wdz[ka * CZ4_ + lo];      b0.y = Wdz[(ka + 1) * CZ4_ + lo];
    v2f b1; b1.x = Wdz[ka * CZ4_ + 16 + lo]; b1.y = Wdz[(ka + 1) * CZ4_ + 16 + lo];
    v2f b2;
    b2.x = (lo < H_) ? Wb[ka * H_ + lo] : 0.0f;
    b2.y = (lo < H_) ? Wb[(ka + 1) * H_ + lo] : 0.0f;
    c0 = wmma4(a, b0, c0);
    c1 = wmma4(a, b1, c1);
    cb = wmma4(a, b2, cb);
  }
  const float bd0 = bdz[lo], bd1 = bdz[16 + lo];
  const float bbv = (lo < H_) ? bb[lo] : 0.0f;
#pragma unroll
  for (int r = 0; r < 8; ++r) {
    const size_t row = rowb + r + 8 * hi;
    pz[row * CZ4_ + lo] = c0[r] + bd0;
    pz[row * CZ4_ + 16 + lo] = c1[r] + bd1;
    if (lo < H_) bias2d[row * H_ + lo] = cb[r] + bbv;
  }
}

// robust Cholesky NLL contribution for one Gaussian pair
__device__ __forceinline__ void nll_pair(const float* __restrict__ qd,
                                         const float* __restrict__ kd,
                                         float& geo) {
  const float d0 = qd[0] - kd[0], d1 = qd[1] - kd[1], d2 = qd[2] - kd[2];
  const float s00 = qd[3] + kd[3], s01 = qd[4] + kd[4], s02 = qd[5] + kd[5];
  const float s11 = qd[6] + kd[6], s12 = qd[7] + kd[7], s22 = qd[8] + kd[8];
  const float eps = 1e-8f;
  const float l00sq = fmaxf(s00, eps);
  const float il00 = __frsqrt_rn(l00sq);
  const float l10 = s01 * il00, l20 = s02 * il00;
  const float l11sq = fmaxf(s11 - l10 * l10, eps);
  const float il11 = __frsqrt_rn(l11sq);
  const float l21 = (s12 - l20 * l10) * il11;
  const float l22sq = fmaxf(s22 - l20 * l20 - l21 * l21, eps);
  const float il22 = __frsqrt_rn(l22sq);
  const float logdet = __logf(l00sq) + __logf(l11sq) + __logf(l22sq);
  const float y0 = d0 * il00;
  const float y1 = (d1 - l10 * y0) * il11;
  const float y2 = (d2 - l20 * y0 - l21 * y1) * il22;
  geo += -0.5f * (y0 * y0 + y1 * y1 + y2 * y2) - 0.5f * logdet;
}

// ---------------------------------------------------------------------------
// Logits + softmax.  One wave per (b,h, 16-row i tile).
// Pass 1: qk^T (WMMA) + pair-bias + geometry + mask -> raw logits to wbuf,
//         running row max via __shfl_xor over the 16-lane row halves.
// Pass 2: p = exp(s - m) back to wbuf, row sums -> linv = 1/sum.
// ---------------------------------------------------------------------------
__global__ __launch_bounds__(32) void iga_logits_softmax(
    const float* __restrict__ qproj, const float* __restrict__ kproj,
    const float* __restrict__ bias2d, const float* __restrict__ qmu,
    const float* __restrict__ qsig, const float* __restrict__ kmu,
    const float* __restrict__ ksig, const float* __restrict__ mask,
    const float* __restrict__ geo_scale, const float* __restrict__ geo_bias,
    float* __restrict__ wbuf, float* __restrict__ linv) {
  const int L = threadIdx.x, lo = L & 15, hi = L >> 4;
  const int it = blockIdx.x % NT16;
  const int h = (blockIdx.x / NT16) % H_;
  const int b = blockIdx.x / (NT16 * H_);
  const int ibase = it * 16;

  __shared__ float sQ[16 * G_ * 9];
  __shared__ float sK[16 * G_ * 9];
  __shared__ float sMi[16], sMj[16];

  for (int idx = L; idx < 16 * G_ * 9; idx += 32) {
    const int i16 = idx / (G_ * 9);
    const int rem = idx % (G_ * 9);
    const int g = rem / 9, c = rem % 9;
    const size_t base = (size_t)((b * N_ + ibase + i16) * H_ + h) * G_ + g;
    sQ[idx] = (c < 3) ? qmu[base * 3 + c] : qsig[base * 6 + (c - 3)];
  }
  if (L < 16) sMi[L] = mask[b * N_ + ibase + L];

  const float ascale = __logf(1.0f + __expf(geo_scale[h]));  // softplus
  const float gbias = geo_bias[h];

  float rm[8];
#pragma unroll
  for (int r = 0; r < 8; ++r) rm[r] = -3.0e38f;

  const float* qrow = qproj + (size_t)(b * N_ + ibase + lo) * HC_ + h * CH_;
  const size_t wrowbase = ((size_t)(b * H_ + h) * N_ + ibase) * N_;

  for (int jt = 0; jt < NT16; ++jt) {
    const int jbase = jt * 16;
    __syncthreads();
    for (int idx = L; idx < 16 * G_ * 9; idx += 32) {
      const int j16 = idx / (G_ * 9);
      const int rem = idx % (G_ * 9);
      const int g = rem / 9, c = rem % 9;
      const size_t base = (size_t)((b * N_ + jbase + j16) * H_ + h) * G_ + g;
      sK[idx] = (c < 3) ? kmu[base * 3 + c] : ksig[base * 6 + (c - 3)];
    }
    if (L < 16) sMj[L] = mask[b * N_ + jbase + L];
    __syncthreads();

    v8f acc = {};
    const float* krow = kproj + (size_t)(b * N_ + jbase + lo) * HC_ + h * CH_;
#pragma unroll
    for (int k = 0; k < CH_; k += 4) {
      const int ka = k + 2 * hi;
      v2f a;  a.x = qrow[ka]; a.y = qrow[ka + 1];
      v2f bf; bf.x = krow[ka]; bf.y = krow[ka + 1];
      acc = wmma4(a, bf, acc);
    }

    const float mj = sMj[lo];
    const int j = jbase + lo;
#pragma unroll
    for (int r = 0; r < 8; ++r) {
      const int i16 = r + 8 * hi;
      const int i = ibase + i16;
      float logit = acc[r] * 0.25f  // 1/sqrt(CH)
          + bias2d[((size_t)(b * N_ + i) * N_ + j) * H_ + h];
      float geo = 0.0f;
#pragma unroll
      for (int g = 0; g < G_; ++g)
        nll_pair(&sQ[(i16 * G_ + g) * 9], &sK[(lo * G_ + g) * 9], geo);
      const float m2d = sMi[i16] * mj;
      logit += ascale * (geo * m2d) + gbias + (1.0f - m2d) * (-100000.0f);
      wbuf[wrowbase + (size_t)i16 * N_ + j] = logit;
      float v = logit;
#pragma unroll
      for (int mm = 1; mm < 16; mm <<= 1)
        v = fmaxf(v, __shfl_xor(v, mm, 32));
      rm[r] = fmaxf(rm[r], v);
    }
  }

  float ssum[8];
#pragma unroll
  for (int r = 0; r < 8; ++r) ssum[r] = 0.0f;
  for (int jt = 0; jt < NT16; ++jt) {
    const int j = jt * 16 + lo;
#pragma unroll
    for (int r = 0; r < 8; ++r) {
      const int i16 = r + 8 * hi;
      const size_t addr = wrowbase + (size_t)i16 * N_ + j;
      const float p = __expf(wbuf[addr] - rm[r]);
      wbuf[addr] = p;
      ssum[r] += p;
    }
  }
#pragma unroll
  for (int r = 0; r < 8; ++r) {
#pragma unroll
    for (int mm = 1; mm < 16; mm <<= 1) ssum[r] += __shfl_xor(ssum[r], mm, 32);
    if (lo == 0)
      linv[(size_t)(b * H_ + h) * N_ + ibase + r + 8 * hi] = 1.0f / ssum[r];
  }
}

// ---------------------------------------------------------------------------
// o = p @ v (16 cols) and opt = p @ vpg (24 cols, padded to 32).
// One wave per (b,h,i-tile); K = N = 384, all via WMMA.
// ---------------------------------------------------------------------------
__global__ __launch_bounds__(32) void iga_attn_ov(
    const float* __restrict__ wbuf, const float* __restrict__ vproj,
    const float* __restrict__ vpg, float* __restrict__ o_un,
    float* __restrict__ opt_un) {
  const int L = threadIdx.x, lo = L & 15, hi = L >> 4;
  const int it = blockIdx.x % NT16;
  const int h = (blockIdx.x / NT16) % H_;
  const int b = blockIdx.x / (NT16 * H_);
  const int ibase = it * 16;
  const float* Arow = wbuf + ((size_t)(b * H_ + h) * N_ + ibase + lo) * N_;
  v8f co = {}, cp0 = {}, cp1 = {};
  for (int k = 0; k < N_; k += 4) {
    const int ka = k + 2 * hi;
    v2f a; a.x = Arow[ka]; a.y = Arow[ka + 1];
    v2f bo;
    bo.x = vproj[(size_t)(b * N_ + ka) * HC_ + h * CH_ + lo];
    bo.y = vproj[(size_t)(b * N_ + ka + 1) * HC_ + h * CH_ + lo];
    co = wmma4(a, bo, co);
    const float* vg0 = vpg + ((size_t)(b * N_ + ka) * H_ + h) * (P_ * 3);
    const float* vg1 = vpg + ((size_t)(b * N_ + ka + 1) * H_ + h) * (P_ * 3);
    v2f b0; b0.x = vg0[lo]; b0.y = vg1[lo];
    v2f b1;
    b1.x = (lo < 8) ? vg0[16 + lo] : 0.0f;
    b1.y = (lo < 8) ? vg1[16 + lo] : 0.0f;
    cp0 = wmma4(a, b0, cp0);
    cp1 = wmma4(a, b1, cp1);
  }
#pragma unroll
  for (int r = 0; r < 8; ++r) {
    const int i = ibase + r + 8 * hi;
    o_un[(size_t)(b * N_ + i) * HC_ + h * CH_ + lo] = co[r];
    float* od = opt_un + ((size_t)(b * N_ + i) * H_ + h) * (P_ * 3);
    od[lo] = cp0[r];
    if (lo < 8) od[16 + lo] = cp1[r];
  }
}

// ---------------------------------------------------------------------------
// opair[b,i,h,:] = sum_j p[b,h,i,j] * pz[b,i,j,:]  (pz depends on i!)
// One wave per (b,i): A = p over h (8 rows padded to 16) x 384, B = pz_i.
// ---------------------------------------------------------------------------
__global__ __launch_bounds__(32) void iga_attn_opair(
    const float* __restrict__ wbuf, const float* __restrict__ pz,
    float* __restrict__ opair_un) {
  const int L = threadIdx.x, lo = L & 15, hi = L >> 4;
  const int bn = blockIdx.x;
  const int b = bn / N_, i = bn % N_;
  const float* Arow =
      (lo < H_) ? wbuf + ((size_t)(b * H_ + lo) * N_ + i) * N_ : nullptr;
  const float* Pz = pz + (size_t)bn * N_ * CZ4_;
  v8f c0 = {}, c1 = {};
  for (int k = 0; k < N_; k += 4) {
    const int ka = k + 2 * hi;
    v2f a;
    a.x = (lo < H_) ? Arow[ka] : 0.0f;
    a.y = (lo < H_) ? Arow[ka + 1] : 0.0f;
    v2f b0; b0.x = Pz[(size_t)ka * CZ4_ + lo];      b0.y = Pz[(size_t)(ka + 1) * CZ4_ + lo];
    v2f b1; b1.x = Pz[(size_t)ka * CZ4_ + 16 + lo]; b1.y = Pz[(size_t)(ka + 1) * CZ4_ + 16 + lo];
    c0 = wmma4(a, b0, c0);
    c1 = wmma4(a, b1, c1);
  }
  if (hi == 0) {  // rows M = 0..7 == heads
#pragma unroll
    for (int r = 0; r < 8; ++r) {
      float* od = opair_un + ((size_t)bn * H_ + r) * CZ4_;
      od[lo] = c0[r];
      od[16 + lo] = c1[r];
    }
  }
}

// ---------------------------------------------------------------------------
// Post: normalize by 1/l, invert frame for points, norms, assemble cat(640).
// One block per (b,i), 64 threads = (h,p).
// ---------------------------------------------------------------------------
__global__ __launch_bounds__(64) void iga_post_cat(
    const float* __restrict__ o_un, const float* __restrict__ opt_un,
    const float* __restrict__ opair_un, const float* __restrict__ linv,
    const float* __restrict__ rot, const float* __restrict__ trans,
    float* __restrict__ cat) {
  const int bn = blockIdx.x;
  const int b = bn / N_, i = bn % N_;
  const int t = threadIdx.x;  // h*8+p
  const int h = t >> 3, p = t & 7;
  const float li = linv[(size_t)(b * H_ + h) * N_ + i];
  const float* R = rot + (size_t)bn * 9;
  const float* tr = trans + (size_t)bn * 3;
  const float* ou = opt_un + ((size_t)bn * H_ + h) * (P_ * 3) + p * 3;
  const float g0 = ou[0] * li - tr[0];
  const float g1 = ou[1] * li - tr[1];
  const float g2 = ou[2] * li - tr[2];
  const float lx = R[0] * g0 + R[3] * g1 + R[6] * g2;  // R^T (g - t)
  const float ly = R[1] * g0 + R[4] * g1 + R[7] * g2;
  const float lz = R[2] * g0 + R[5] * g1 + R[8] * g2;
  float* C = cat + (size_t)bn * CATD;
  C[128 + t * 3 + 0] = lx;
  C[128 + t * 3 + 1] = ly;
  C[128 + t * 3 + 2] = lz;
  C[320 + t] = sqrtf(lx * lx + ly * ly + lz * lz + 1e-8f);
#pragma unroll
  for (int e2 = 0; e2 < 2; ++e2) {
    const int e = 2 * t + e2;
    C[e] = o_un[(size_t)bn * HC_ + e] *
           linv[(size_t)(b * H_ + (e >> 4)) * N_ + i];
  }
#pragma unroll
  for (int e4 = 0; e4 < 4; ++e4) {
    const int e = 4 * t + e4;
    C[384 + e] = opair_un[(size_t)bn * (H_ * CZ4_) + e] *
                 linv[(size_t)(b * H_ + (e >> 5)) * N_ + i];
  }
}

// ---------------------------------------------------------------------------
extern "C" void kernel_launch(void* const* d_in, const int* in_sizes, int n_in,
                              void* d_out, int out_size, void* d_ws,
                              size_t ws_size, hipStream_t stream) {
  const float* s     = (const float*)d_in[0];
  const float* z     = (const float*)d_in[1];
  const float* mask  = (const float*)d_in[2];
  const float* rot   = (const float*)d_in[3];
  const float* trans = (const float*)d_in[4];
  const float* lmean = (const float*)d_in[5];
  const float* slog  = (const float*)d_in[6];
  const float* Wq    = (const float*)d_in[7];
  const float* Wk    = (const float*)d_in[8];
  const float* Wv    = (const float*)d_in[9];
  const float* Wb    = (const float*)d_in[10];
  const float* bb    = (const float*)d_in[11];
  const float* Wdz   = (const float*)d_in[12];
  const float* bdz   = (const float*)d_in[13];
  const float* Wqg   = (const float*)d_in[14];
  const float* bqg   = (const float*)d_in[15];
  const float* Wkg   = (const float*)d_in[16];
  const float* bkg   = (const float*)d_in[17];
  const float* Wvp   = (const float*)d_in[18];
  const float* bvp   = (const float*)d_in[19];
  const float* gsc   = (const float*)d_in[20];
  const float* gbi   = (const float*)d_in[21];
  const float* Wout  = (const float*)d_in[22];
  const float* bout  = (const float*)d_in[23];
  float* out = (float*)d_out;

  const size_t BN = (size_t)B_ * N_;        // 768
  const size_t BNN = BN * N_;               // 294912
  const size_t BNHG = BN * H_ * G_;         // 12288

  float* ws = (float*)d_ws;
  size_t off = 0;
  auto alloc = [&](size_t n) { float* p = ws + off; off += n; return p; };
  float* qproj    = alloc(BN * HC_);
  float* kproj    = alloc(BN * HC_);
  float* vproj    = alloc(BN * HC_);
  float* qgfeat   = alloc(BN * H_ * G_ * 6);
  float* kgfeat   = alloc(BN * H_ * G_ * 6);
  float* vpts     = alloc(BN * H_ * P_ * 3);
  float* qmu      = alloc(BNHG * 3);
  float* qsig     = alloc(BNHG * 6);
  float* kmu      = alloc(BNHG * 3);
  float* ksig     = alloc(BNHG * 6);
  float* vpg      = alloc(BN * H_ * P_ * 3);
  float* pz       = alloc(BNN * CZ4_);       // 37.7 MB
  float* bias2d   = alloc(BNN * H_);         // 9.4 MB
  float* wbuf     = alloc((size_t)B_ * H_ * N_ * N_);  // 9.4 MB
  float* linvb    = alloc((size_t)B_ * H_ * N_);
  float* o_un     = alloc(BN * HC_);
  float* opt_un   = alloc(BN * H_ * P_ * 3);
  float* opair_un = alloc(BN * H_ * CZ4_);
  float* cat      = alloc(BN * CATD);        // ~65 MB total

  // s-projections (all f32 WMMA GEMMs)
  gemm_f32_wmma<<<dim3(48, 8), 32, 0, stream>>>(s, Wq, nullptr, qproj, 768, 384, 128);
  gemm_f32_wmma<<<dim3(48, 8), 32, 0, stream>>>(s, Wk, nullptr, kproj, 768, 384, 128);
  gemm_f32_wmma<<<dim3(48, 8), 32, 0, stream>>>(s, Wv, nullptr, vproj, 768, 384, 128);
  gemm_f32_wmma<<<dim3(48, 6), 32, 0, stream>>>(s, Wqg, bqg, qgfeat, 768, 384, 96);
  gemm_f32_wmma<<<dim3(48, 6), 32, 0, stream>>>(s, Wkg, bkg, kgfeat, 768, 384, 96);
  gemm_f32_wmma<<<dim3(48, 12), 32, 0, stream>>>(s, Wvp, bvp, vpts, 768, 384, 192);

  iga_geom_prep<<<(int)((BNHG + 127) / 128), 128, 0, stream>>>(
      qgfeat, kgfeat, lmean, slog, rot, trans, qmu, qsig, kmu, ksig);
  iga_vpg_prep<<<(int)((BN * H_ * P_ + 127) / 128), 128, 0, stream>>>(
      vpts, rot, trans, vpg);

  // single streaming pass over z: pz = z@Wdz+bdz, bias2d = z@Wb+bb
  iga_zproj<<<(int)(BNN / 16), 32, 0, stream>>>(z, Wdz, bdz, Wb, bb, pz, bias2d);

  // logits + softmax (unnormalized p -> wbuf, 1/rowsum -> linv)
  iga_logits_softmax<<<B_ * H_ * NT16, 32, 0, stream>>>(
      qproj, kproj, bias2d, qmu, qsig, kmu, ksig, mask, gsc, gbi, wbuf, linvb);

  iga_attn_ov<<<B_ * H_ * NT16, 32, 0, stream>>>(wbuf, vproj, vpg, o_un, opt_un);
  iga_attn_opair<<<(int)BN, 32, 0, stream>>>(wbuf, pz, opair_un);
  iga_post_cat<<<(int)BN, 64, 0, stream>>>(o_un, opt_un, opair_un, linvb, rot,
                                           trans, cat);

  // final projection
  gemm_f32_wmma<<<dim3(48, 24), 32, 0, stream>>>(cat, Wout, bout, out, 768, 640, 384);
}